// GAT_23046794510609
// MI455X (gfx1250) — compile-verified
//
#include <hip/hip_runtime.h>
#include <math.h>

typedef float v2f __attribute__((ext_vector_type(2)));
typedef float v8f __attribute__((ext_vector_type(8)));

#define NEG_SLOPE 0.2f

// ---- ordered-uint encoding for float atomic max (monotone bijection) ----
static __device__ __forceinline__ unsigned f2ord(float f) {
    unsigned u = __float_as_uint(f);
    return (u & 0x80000000u) ? ~u : (u | 0x80000000u);
}
static __device__ __forceinline__ float ord2f(unsigned u) {
    return __uint_as_float((u & 0x80000000u) ? (u ^ 0x80000000u) : ~u);
}

// -------- init per-layer segment state: m = -inf (ord), sum = 0, acc = 0 ----
__global__ __launch_bounds__(256) void gat_seg_init(unsigned* __restrict__ mo,
                                                    float* __restrict__ ssum,
                                                    float* __restrict__ acc,
                                                    int N, long long totF) {
    long long i = (long long)blockIdx.x * blockDim.x + threadIdx.x;
    if (i < N) { mo[i] = 0x007FFFFFu; /* f2ord(-inf) */ ssum[i] = 0.0f; }
    if (i < totF) acc[i] = 0.0f;
}

// -------- H[N x F] = X[N x K] @ W[K x F] via V_WMMA_F32_16X16X4_F32 --------
// Block = 8 waves; each wave computes one 16x16 tile (8 row-tiles x 1 col-tile).
// B slice (K x 16) staged in LDS once per block, zero-padded, interleaved as
// (k, k+1) pairs so each lane's B fragment is one conflict-free ds_load_b64:
//   element (k, c) lives at word  ((k>>1)*16 + c)*2 + (k&1)
template <int K, int F>
__global__ __launch_bounds__(256) void gat_wmma_linear(const float* __restrict__ X,
                                                       const float* __restrict__ W,
                                                       float* __restrict__ H,
                                                       int N) {
    __shared__ float Bs[K * 16];
    const int tid  = threadIdx.x;
    const int col0 = blockIdx.y * 16;

    // cooperative, zero-padded staging of W[:, col0:col0+16]
    for (int idx = tid; idx < K * 16; idx += 256) {
        const int k = idx >> 4;
        const int c = idx & 15;
        const int col = col0 + c;
        const float v = (col < F) ? W[k * F + col] : 0.0f;
        Bs[(((k >> 1) * 16 + c) << 1) + (k & 1)] = v;
    }
    __syncthreads();

    const int wave = tid >> 5;
    const int lane = tid & 31;
    const int row0 = (blockIdx.x * 8 + wave) * 16;
    if (row0 >= N) return;                 // uniform per wave

    const int half = lane >> 4;            // 0: lanes 0-15, 1: lanes 16-31
    const int l16  = lane & 15;
    int arow = row0 + l16;
    if (arow >= N) arow = N - 1;           // clamp: OOB rows computed but never stored
    // A-frag VGPR0 -> K = k0 + 2*half, VGPR1 -> K+1 (8B aligned: K % 4 == 0)
    const float* xrow  = X + (size_t)arow * K + 2 * half;
    const v2f*   bbase = (const v2f*)Bs + (half * 16 + l16);

    v8f c = {};
#pragma unroll
    for (int k0 = 0; k0 < K; k0 += 4) {
        const v2f a = *(const v2f*)(xrow + k0);
        const v2f b = bbase[(k0 >> 1) * 16];   // pair (k0/2 + half), column l16
        // (neg_a, A, neg_b, B, c_mod, C, reuse_a, reuse_b)
        c = __builtin_amdgcn_wmma_f32_16x16x4_f32(false, a, false, b, (short)0, c,
                                                  false, false);
    }

    // D layout: VGPR j -> M = j + 8*half, N = lane&15
    const int bcol = col0 + l16;
#pragma unroll
    for (int j = 0; j < 8; ++j) {
        const int r = row0 + j + 8 * half;
        if (r < N && bcol < F) H[(size_t)r * F + bcol] = c[j];
    }
}

// -------- per-node attention scalars: ssrc = h.a_src, sdst = h.a_dst --------
template <int F>
__global__ __launch_bounds__(256) void gat_node_att(const float* __restrict__ H,
                                                    const float* __restrict__ a_src,
                                                    const float* __restrict__ a_dst,
                                                    float* __restrict__ ssrc,
                                                    float* __restrict__ sdst,
                                                    int N) {
    const int n = blockIdx.x * blockDim.x + threadIdx.x;
    if (n >= N) return;
    const float4* h4 = (const float4*)(H + (size_t)n * F);
    float s0 = 0.0f, s1 = 0.0f;
#pragma unroll
    for (int q = 0; q < F / 4; ++q) {
        const float4 v  = h4[q];
        const float4 as = ((const float4*)a_src)[q];
        const float4 ad = ((const float4*)a_dst)[q];
        s0 += v.x * as.x + v.y * as.y + v.z * as.z + v.w * as.w;
        s1 += v.x * ad.x + v.y * ad.y + v.z * ad.z + v.w * ad.w;
    }
    ssrc[n] = s0;
    sdst[n] = s1;
}

static __device__ __forceinline__ void edge_sd(const int* __restrict__ ei, int E,
                                               int e, int* s, int* d) {
    if (e < E) { *s = ei[e]; *d = ei[E + e]; }
    else       { *s = *d = e - E; }        // self-loop edges appended
}

// -------- pass 1: logits + segment max --------
__global__ __launch_bounds__(256) void gat_edge_logits(const int* __restrict__ ei,
                                                       int E, int N,
                                                       const float* __restrict__ ssrc,
                                                       const float* __restrict__ sdst,
                                                       float* __restrict__ elog,
                                                       unsigned* __restrict__ mo) {
    const int e = blockIdx.x * blockDim.x + threadIdx.x;
    if (e >= E + N) return;
    int s, d; edge_sd(ei, E, e, &s, &d);
    float v = ssrc[s] + sdst[d];
    v = (v > 0.0f) ? v : NEG_SLOPE * v;    // LeakyReLU(0.2)
    elog[e] = v;
    atomicMax(mo + d, f2ord(v));
}

// -------- pass 2: exp(e - m[dst]) + segment sum --------
__global__ __launch_bounds__(256) void gat_edge_exp(const int* __restrict__ ei,
                                                    int E, int N,
                                                    float* __restrict__ elog,
                                                    const unsigned* __restrict__ mo,
                                                    float* __restrict__ ssum) {
    const int e = blockIdx.x * blockDim.x + threadIdx.x;
    if (e >= E + N) return;
    int s, d; edge_sd(ei, E, e, &s, &d);
    const float ex = __expf(elog[e] - ord2f(mo[d]));
    elog[e] = ex;
    atomicAdd(ssum + d, ex);
}

// -------- pass 3: acc[dst] += h[src] * alpha --------
// 64 threads per edge (padded), f = t & 63: no integer division anywhere.
template <int F>
__global__ __launch_bounds__(256) void gat_edge_aggregate(const int* __restrict__ ei,
                                                          int E, int N,
                                                          const float* __restrict__ elog,
                                                          const float* __restrict__ ssum,
                                                          const float* __restrict__ H,
                                                          float* __restrict__ acc) {
    const long long t = (long long)blockIdx.x * blockDim.x + threadIdx.x;
    const int e = (int)(t >> 6);
    const int f = (int)(t & 63);
    if (e >= E + N || f >= F) return;
    int s, d; edge_sd(ei, E, e, &s, &d);
    const float alpha = elog[e] / ssum[d];
    atomicAdd(acc + (size_t)d * F + f, H[(size_t)s * F + f] * alpha);
}

// -------- finalize: ReLU(acc + b) for layers 1..3 --------
template <int F>
__global__ __launch_bounds__(256) void gat_finalize_relu(const float* __restrict__ acc,
                                                         const float* __restrict__ b,
                                                         float* __restrict__ out,
                                                         long long NF) {
    const long long t = (long long)blockIdx.x * blockDim.x + threadIdx.x;
    if (t >= NF) return;
    const int f = (int)(t % F);
    const float v = acc[t] + b[f];
    out[t] = (v > 0.0f) ? v : 0.0f;
}

// -------- finalize: row softmax(acc + b) for the last layer --------
template <int F>
__global__ __launch_bounds__(256) void gat_finalize_softmax(const float* __restrict__ acc,
                                                            const float* __restrict__ b,
                                                            float* __restrict__ out,
                                                            int N) {
    const int n = blockIdx.x * blockDim.x + threadIdx.x;
    if (n >= N) return;
    const float* a = acc + (size_t)n * F;
    float tmp[F];
    float mx = -INFINITY;
#pragma unroll
    for (int f = 0; f < F; ++f) {
        const float v = a[f] + b[f];
        tmp[f] = v;
        mx = fmaxf(mx, v);
    }
    float s = 0.0f;
#pragma unroll
    for (int f = 0; f < F; ++f) {
        const float ex = __expf(tmp[f] - mx);
        tmp[f] = ex;
        s += ex;
    }
    const float inv = 1.0f / s;
    float* o = out + (size_t)n * F;
#pragma unroll
    for (int f = 0; f < F; ++f) o[f] = tmp[f] * inv;
}

extern "C" void kernel_launch(void* const* d_in, const int* in_sizes, int n_in,
                              void* d_out, int out_size, void* d_ws, size_t ws_size,
                              hipStream_t stream) {
    const int D_IN = 128, HID = 64, OUTF = 40;
    const int N = in_sizes[0] / D_IN;        // 100000
    const int E = in_sizes[1] / 2;           // 1600000

    const float* x  = (const float*)d_in[0];
    const int*   ei = (const int*)d_in[1];
    const float* W[4]    = {(const float*)d_in[2],  (const float*)d_in[6],
                            (const float*)d_in[10], (const float*)d_in[14]};
    const float* asrc[4] = {(const float*)d_in[3],  (const float*)d_in[7],
                            (const float*)d_in[11], (const float*)d_in[15]};
    const float* adst[4] = {(const float*)d_in[4],  (const float*)d_in[8],
                            (const float*)d_in[12], (const float*)d_in[16]};
    const float* bias[4] = {(const float*)d_in[5],  (const float*)d_in[9],
                            (const float*)d_in[13], (const float*)d_in[17]};

    // workspace carve-up (all f32-sized)
    float* ws     = (float*)d_ws;
    float* h      = ws;              ws += (size_t)N * HID;   // transformed features
    float* acc    = ws;              ws += (size_t)N * HID;   // scatter accumulator
    float* featin = ws;              ws += (size_t)N * HID;   // inter-layer activations
    float* ssrc   = ws;              ws += N;
    float* sdst   = ws;              ws += N;
    float* ssum   = ws;              ws += N;
    unsigned* mo  = (unsigned*)ws;   ws += N;
    float* elog   = ws;                                        // E+N per-edge scratch

    const int E2 = E + N;
    const int TB = 256;
    auto cdiv = [](long long a, long long b) { return (unsigned)((a + b - 1) / b); };
    const unsigned rowBlocks = cdiv((N + 15) / 16, 8);

    const float* in = x;
    for (int L = 0; L < 4; ++L) {
        const int F = (L == 3) ? OUTF : HID;
        const long long NF = (long long)N * F;

        gat_seg_init<<<cdiv(NF, TB), TB, 0, stream>>>(mo, ssum, acc, N, NF);

        if (L == 0)
            gat_wmma_linear<128, 64><<<dim3(rowBlocks, 4), 256, 0, stream>>>(in, W[L], h, N);
        else if (L < 3)
            gat_wmma_linear<64, 64><<<dim3(rowBlocks, 4), 256, 0, stream>>>(in, W[L], h, N);
        else
            gat_wmma_linear<64, 40><<<dim3(rowBlocks, 3), 256, 0, stream>>>(in, W[L], h, N);

        if (L < 3)
            gat_node_att<64><<<cdiv(N, TB), TB, 0, stream>>>(h, asrc[L], adst[L], ssrc, sdst, N);
        else
            gat_node_att<40><<<cdiv(N, TB), TB, 0, stream>>>(h, asrc[L], adst[L], ssrc, sdst, N);

        gat_edge_logits<<<cdiv(E2, TB), TB, 0, stream>>>(ei, E, N, ssrc, sdst, elog, mo);
        gat_edge_exp<<<cdiv(E2, TB), TB, 0, stream>>>(ei, E, N, elog, mo, ssum);

        if (L < 3)
            gat_edge_aggregate<64><<<cdiv((long long)E2 * 64, TB), TB, 0, stream>>>(
                ei, E, N, elog, ssum, h, acc);
        else
            gat_edge_aggregate<40><<<cdiv((long long)E2 * 64, TB), TB, 0, stream>>>(
                ei, E, N, elog, ssum, h, acc);

        if (L < 3) {
            gat_finalize_relu<64><<<cdiv(NF, TB), TB, 0, stream>>>(acc, bias[L], featin, NF);
            in = featin;
        } else {
            gat_finalize_softmax<40><<<cdiv(N, TB), TB, 0, stream>>>(acc, bias[L],
                                                                     (float*)d_out, N);
        }
    }
}